// Net_15685220565735
// MI455X (gfx1250) — compile-verified
//
#include <hip/hip_runtime.h>
#include <math.h>

// ---------------------------------------------------------------------------
// CDNA5 (gfx1250) INRF network.
//   out = (logits[256*10], x00[256*32*28*28], x01[256*32*28*28])
// Layers 2/3 run on V_WMMA_F32_16X16X4_F32 (fp32 matrix pipe, full precision).
// ---------------------------------------------------------------------------

typedef __attribute__((ext_vector_type(2))) float v2f;
typedef __attribute__((ext_vector_type(8))) float v8f;

__device__ __forceinline__ v8f wmma_k4(v2f a, v2f b, v8f c) {
  // D(16x16,f32) = A(16x4,f32) * B(4x16,f32) + C
  return __builtin_amdgcn_wmma_f32_16x16x4_f32(false, a, false, b, (short)0, c,
                                               false, false);
}

// Cephes-style odd minimax polynomial for atan, |z| <= ~0.66 (rel err ~2e-7).
// Here |z| = sigma*|x_shift - x_center| <= ~0.45 (layer1) / ~0.1 (layers 2,3),
// so the libm large-argument path (rcp + selects) is dead weight; this is
// 4 FMA + 2 mul and packs into v_pk_fma_f32.
__device__ __forceinline__ float fast_atan(float z) {
  float t = z * z;
  float p = fmaf(t, 8.05374449538e-2f, -1.38776856032e-1f);
  p = fmaf(t, p, 1.99777106478e-1f);
  p = fmaf(t, p, -3.33329491539e-1f);
  return fmaf(z * t, p, z);
}

// ---------------------------------------------------------------------------
// Layer 1: INRF 1->32 on 28x28, sigma=0.05.  C_in==1 so matrix cores are
// wasted here; plain VALU with the padded image + weights in LDS.
// One block per image, 256 threads.
// ---------------------------------------------------------------------------
__global__ void __launch_bounds__(256) inrf1_kernel(
    const float* __restrict__ x, const float* __restrict__ m1,
    const float* __restrict__ w1, const float* __restrict__ b1,
    float* __restrict__ x00) {
  __shared__ float xp[32 * 33];  // padded 32x32 image, row stride 33
  __shared__ float mw[800];      // m1[o][j]
  __shared__ float wn[800];      // -w1[o][j]
  const int b = blockIdx.x, tid = threadIdx.x;
  for (int i = tid; i < 32 * 33; i += 256) xp[i] = 0.0f;
  for (int i = tid; i < 800; i += 256) {
    mw[i] = m1[i];
    wn[i] = -w1[i];
  }
  __syncthreads();
  const float* xb = x + (size_t)b * 784;
  for (int i = tid; i < 784; i += 256) {
    int h = i / 28, w = i - h * 28;
    xp[(h + 2) * 33 + (w + 2)] = xb[i];
  }
  __syncthreads();
  float* ob = x00 + (size_t)b * 32 * 784;
  for (int p = tid; p < 784; p += 256) {
    int h = p / 28, w = p - h * 28;
    float xc = xp[(h + 2) * 33 + (w + 2)];
    float acc[32];
#pragma unroll
    for (int o = 0; o < 32; o++) acc[o] = b1[o];
    for (int ky = 0; ky < 5; ky++)
      for (int kx = 0; kx < 5; kx++) {
        float s = xp[(h + ky) * 33 + (w + kx)];
        float d = fast_atan(0.05f * (s - xc));
        int j = ky * 5 + kx;
#pragma unroll
        for (int o = 0; o < 32; o++)
          acc[o] = fmaf(mw[o * 25 + j], s, fmaf(wn[o * 25 + j], d, acc[o]));
      }
#pragma unroll
    for (int o = 0; o < 32; o++) ob[o * 784 + p] = acc[o];
  }
}

// ---------------------------------------------------------------------------
// Layer 2: INRF 32->32 on 28x28, sigma=0.01.  WMMA f32 16x16x4.
// Grid: 512 blocks = (image, row-half).  Each block stages a zero-padded
// 32ch x 18row x 33 LDS halo tile (rows r0-2 .. r0+15) and loops the 25
// offsets outermost, staging that offset's (32x32) m / -w weight slices in
// LDS.  Each wave owns up to 4 tiles of 16 pixels; accumulators (16out x
// 16pix, two out-halves) stay resident in VGPRs across all 25 offsets.
// ---------------------------------------------------------------------------
__global__ void __launch_bounds__(256) inrf2_kernel(
    const float* __restrict__ xin, const float* __restrict__ m2,
    const float* __restrict__ w2, const float* __restrict__ b2,
    float* __restrict__ xout) {
  extern __shared__ float smem[];
  float* xp = smem;                  // 32*18*33 = 19008 floats
  float* mw = smem + 32 * 18 * 33;   // 32*33 = 1056
  float* wn = mw + 32 * 33;          // 32*33 = 1056
  const int b = blockIdx.x >> 1, bh = blockIdx.x & 1;
  const int tid = threadIdx.x, lane = tid & 31;
  // SGPR wave id -> tile-validity guards become scalar branches (no exec
  // masking around the WMMA blocks; EXEC provably all-ones at every WMMA).
  const int wave = __builtin_amdgcn_readfirstlane(tid >> 5);
  const int r0 = bh * 14;  // first output row of this half

  for (int i = tid; i < 32 * 18 * 33; i += 256) xp[i] = 0.0f;
  __syncthreads();
  const float* xb = xin + (size_t)b * 32 * 784;
  for (int i = tid; i < 32 * 18 * 28; i += 256) {
    int c = i / (18 * 28), rem = i - c * (18 * 28);
    int lr = rem / 28, w = rem - lr * 28;
    int r = r0 - 2 + lr;
    if (r >= 0 && r < 28) xp[(c * 18 + lr) * 33 + (w + 2)] = xb[c * 784 + r * 28 + w];
  }

  const int col = lane & 15;             // tile column (pixel)
  const int khalf = (lane >> 4) << 1;    // K-pair select: lanes>=16 hold K=2,3
  const int row = lane & 15;             // A row (out channel within half)
  const int rbase = (lane >> 4) * 8;     // D rows r / r+8 split
  int hh[4], ww[4], pp[4];
#pragma unroll
  for (int t = 0; t < 4; t++) {
    int tile = wave + 8 * t;
    int p = tile * 16 + col;  // local position within 14x28 strip
    pp[t] = p;
    if (p > 391) p = 391;     // clamp reads for the partial tile
    hh[t] = p / 28;
    ww[t] = p - hh[t] * 28;
  }
  v8f acc[4][2];
#pragma unroll
  for (int t = 0; t < 4; t++)
#pragma unroll
    for (int u = 0; u < 2; u++)
      acc[t][u] = (v8f){0.f, 0.f, 0.f, 0.f, 0.f, 0.f, 0.f, 0.f};
  __syncthreads();

  for (int off = 0; off < 25; off++) {
    int ky = off / 5, kx = off - ky * 5;
    for (int i = tid; i < 1024; i += 256) {  // stage weight slice (stride 33)
      int o = i >> 5, c = i & 31;
      mw[o * 33 + c] = m2[(size_t)i * 25 + off];
      wn[o * 33 + c] = -w2[(size_t)i * 25 + off];
    }
    __syncthreads();
    for (int kc = 0; kc < 8; kc++) {  // K chunks of 4 over C_in=32
      const int c0 = kc * 4 + khalf, c1 = c0 + 1;
      v2f am0, aw0, am1, aw1;
      am0.x = mw[row * 33 + c0];        am0.y = mw[row * 33 + c1];
      aw0.x = wn[row * 33 + c0];        aw0.y = wn[row * 33 + c1];
      am1.x = mw[(row + 16) * 33 + c0]; am1.y = mw[(row + 16) * 33 + c1];
      aw1.x = wn[(row + 16) * 33 + c0]; aw1.y = wn[(row + 16) * 33 + c1];
#pragma unroll
      for (int t = 0; t < 4; t++) {
        int tile = wave + 8 * t;  // SGPR -> scalar branch
        if (tile < 25) {
          float s0 = xp[(c0 * 18 + hh[t] + ky) * 33 + ww[t] + kx];
          float s1 = xp[(c1 * 18 + hh[t] + ky) * 33 + ww[t] + kx];
          float xc0 = xp[(c0 * 18 + hh[t] + 2) * 33 + ww[t] + 2];
          float xc1 = xp[(c1 * 18 + hh[t] + 2) * 33 + ww[t] + 2];
          v2f bm = {s0, s1};
          v2f bt = {fast_atan(0.01f * (s0 - xc0)), fast_atan(0.01f * (s1 - xc1))};
          acc[t][0] = wmma_k4(am0, bm, acc[t][0]);
          acc[t][0] = wmma_k4(aw0, bt, acc[t][0]);
          acc[t][1] = wmma_k4(am1, bm, acc[t][1]);
          acc[t][1] = wmma_k4(aw1, bt, acc[t][1]);
        }
      }
    }
    __syncthreads();
  }

  float bias[2][8];
#pragma unroll
  for (int u = 0; u < 2; u++)
#pragma unroll
    for (int r = 0; r < 8; r++) bias[u][r] = b2[u * 16 + rbase + r];

  float* ob = xout + (size_t)b * 32 * 784;
#pragma unroll
  for (int t = 0; t < 4; t++) {
    int tile = wave + 8 * t;
    if (tile < 25 && pp[t] < 392) {
      int pg = (r0 + hh[t]) * 28 + ww[t];  // global pixel
#pragma unroll
      for (int u = 0; u < 2; u++)
#pragma unroll
        for (int r = 0; r < 8; r++) {
          int o = u * 16 + rbase + r;
          ob[o * 784 + pg] = acc[t][u][r] + bias[u][r];
        }
    }
  }
}

// ---------------------------------------------------------------------------
// 2x2 maxpool, stride 2.  in: (NC, Hi, Wi) -> out: (NC, Hi/2, Wi/2)
// ---------------------------------------------------------------------------
__global__ void maxpool2_kernel(const float* __restrict__ in,
                                float* __restrict__ out, int NC, int Hi, int Wi) {
  int Ho = Hi >> 1, Wo = Wi >> 1;
  int total = NC * Ho * Wo;
  for (int e = blockIdx.x * blockDim.x + threadIdx.x; e < total;
       e += gridDim.x * blockDim.x) {
    int wo = e % Wo;
    int t = e / Wo;
    int ho = t % Ho;
    int nc = t / Ho;
    const float* p = in + ((size_t)nc * Hi + 2 * ho) * Wi + 2 * wo;
    out[e] = fmaxf(fmaxf(p[0], p[1]), fmaxf(p[Wi], p[Wi + 1]));
  }
}

// ---------------------------------------------------------------------------
// Layer 3: INRF 32->128 on 14x14, sigma=0.01.  WMMA f32 16x16x4.
// One block per image.  LDS: padded 32ch x 18 x 19 input + per-offset
// (128x32) weight slices.  13 pixel-tiles; each wave owns <=2 tiles with
// 8 out-channel halves (128 acc VGPRs) resident across offsets.
// ---------------------------------------------------------------------------
__global__ void __launch_bounds__(256) inrf3_kernel(
    const float* __restrict__ xin, const float* __restrict__ m3,
    const float* __restrict__ w3, const float* __restrict__ b3,
    float* __restrict__ xout) {
  extern __shared__ float smem[];
  float* xp = smem;                  // 32*18*19 = 10944 floats
  float* mw = smem + 32 * 18 * 19;   // 128*33 = 4224
  float* wn = mw + 128 * 33;         // 128*33 = 4224
  const int b = blockIdx.x, tid = threadIdx.x;
  const int lane = tid & 31;
  const int wave = __builtin_amdgcn_readfirstlane(tid >> 5);  // SGPR wave id

  for (int i = tid; i < 32 * 18 * 19; i += 256) xp[i] = 0.0f;
  __syncthreads();
  const float* xb = xin + (size_t)b * 32 * 196;
  for (int i = tid; i < 32 * 196; i += 256) {
    int c = i / 196, p = i - c * 196;
    int h = p / 14, w = p - h * 14;
    xp[(c * 18 + h + 2) * 19 + (w + 2)] = xb[i];
  }

  const int col = lane & 15;
  const int khalf = (lane >> 4) << 1;
  const int row = lane & 15;
  const int rbase = (lane >> 4) * 8;
  int hh[2], ww[2], pp[2];
#pragma unroll
  for (int t = 0; t < 2; t++) {
    int tile = wave + 8 * t;
    int p = tile * 16 + col;
    pp[t] = p;
    if (p > 195) p = 195;
    hh[t] = p / 14;
    ww[t] = p - hh[t] * 14;
  }
  v8f acc[2][8];
#pragma unroll
  for (int t = 0; t < 2; t++)
#pragma unroll
    for (int u = 0; u < 8; u++)
      acc[t][u] = (v8f){0.f, 0.f, 0.f, 0.f, 0.f, 0.f, 0.f, 0.f};
  __syncthreads();

  for (int off = 0; off < 25; off++) {
    int ky = off / 5, kx = off - ky * 5;
    for (int i = tid; i < 4096; i += 256) {
      int o = i >> 5, c = i & 31;
      mw[o * 33 + c] = m3[(size_t)i * 25 + off];
      wn[o * 33 + c] = -w3[(size_t)i * 25 + off];
    }
    __syncthreads();
    for (int kc = 0; kc < 8; kc++) {
      const int c0 = kc * 4 + khalf, c1 = c0 + 1;
      v2f bm[2], bt[2];
#pragma unroll
      for (int t = 0; t < 2; t++) {
        int tile = wave + 8 * t;  // scalar condition
        if (tile < 13) {
          float s0 = xp[(c0 * 18 + hh[t] + ky) * 19 + ww[t] + kx];
          float s1 = xp[(c1 * 18 + hh[t] + ky) * 19 + ww[t] + kx];
          float xc0 = xp[(c0 * 18 + hh[t] + 2) * 19 + ww[t] + 2];
          float xc1 = xp[(c1 * 18 + hh[t] + 2) * 19 + ww[t] + 2];
          bm[t] = (v2f){s0, s1};
          bt[t] = (v2f){fast_atan(0.01f * (s0 - xc0)),
                        fast_atan(0.01f * (s1 - xc1))};
        } else {
          bm[t] = (v2f){0.f, 0.f};
          bt[t] = (v2f){0.f, 0.f};
        }
      }
#pragma unroll
      for (int u = 0; u < 8; u++) {  // 8 out-channel halves (128 channels)
        int o = u * 16 + row;
        v2f am, aw;
        am.x = mw[o * 33 + c0]; am.y = mw[o * 33 + c1];
        aw.x = wn[o * 33 + c0]; aw.y = wn[o * 33 + c1];
#pragma unroll
        for (int t = 0; t < 2; t++) {
          int tile = wave + 8 * t;
          if (tile < 13) {
            acc[t][u] = wmma_k4(am, bm[t], acc[t][u]);
            acc[t][u] = wmma_k4(aw, bt[t], acc[t][u]);
          }
        }
      }
    }
    __syncthreads();
  }

  float bias[8][8];
#pragma unroll
  for (int u = 0; u < 8; u++)
#pragma unroll
    for (int r = 0; r < 8; r++) bias[u][r] = b3[u * 16 + rbase + r];

  float* ob = xout + (size_t)b * 128 * 196;
#pragma unroll
  for (int t = 0; t < 2; t++) {
    int tile = wave + 8 * t;
    if (tile < 13 && pp[t] < 196) {
#pragma unroll
      for (int u = 0; u < 8; u++)
#pragma unroll
        for (int r = 0; r < 8; r++) {
          int o = u * 16 + rbase + r;
          ob[o * 196 + pp[t]] = acc[t][u][r] + bias[u][r];
        }
    }
  }
}

// ---------------------------------------------------------------------------
// Tail: maxpool2(14->7) + avgpool7 + FC(128->10).  One block per image,
// 128 threads (one per feature channel).
// ---------------------------------------------------------------------------
__global__ void __launch_bounds__(128) tail_kernel(
    const float* __restrict__ x02, const float* __restrict__ fcw,
    const float* __restrict__ fcb, float* __restrict__ logits) {
  __shared__ float feat[128];
  const int b = blockIdx.x, tid = threadIdx.x;
  const float* xb = x02 + ((size_t)b * 128 + tid) * 196;
  float s = 0.0f;
  for (int ho = 0; ho < 7; ho++)
    for (int wo = 0; wo < 7; wo++) {
      const float* p = xb + (2 * ho) * 14 + 2 * wo;
      s += fmaxf(fmaxf(p[0], p[1]), fmaxf(p[14], p[15]));
    }
  feat[tid] = s * (1.0f / 49.0f);
  __syncthreads();
  if (tid < 10) {
    float a = fcb[tid];
    for (int o = 0; o < 128; o++) a = fmaf(fcw[tid * 128 + o], feat[o], a);
    logits[b * 10 + tid] = a;
  }
}

// ---------------------------------------------------------------------------
extern "C" void kernel_launch(void* const* d_in, const int* in_sizes, int n_in,
                              void* d_out, int out_size, void* d_ws,
                              size_t ws_size, hipStream_t stream) {
  (void)in_sizes; (void)n_in; (void)out_size; (void)ws_size;
  const float* x   = (const float*)d_in[0];
  const float* m1  = (const float*)d_in[1];
  const float* w1  = (const float*)d_in[2];
  const float* b1  = (const float*)d_in[3];
  const float* m2  = (const float*)d_in[4];
  const float* w2  = (const float*)d_in[5];
  const float* b2  = (const float*)d_in[6];
  const float* m3  = (const float*)d_in[7];
  const float* w3  = (const float*)d_in[8];
  const float* b3  = (const float*)d_in[9];
  const float* fcw = (const float*)d_in[10];
  const float* fcb = (const float*)d_in[11];

  float* out    = (float*)d_out;
  float* logits = out;                 // 256*10
  float* x00    = out + 2560;          // 256*32*28*28 = 6,422,528
  float* x01    = out + 6425088;       // 256*32*28*28

  float* ws  = (float*)d_ws;
  float* x1  = ws;                     // 256*32*14*14 = 1,605,632
  float* x02 = ws + 1605632;           // 256*128*14*14 = 6,422,528

  // Layer 1 (VALU): x -> x00
  inrf1_kernel<<<256, 256, 0, stream>>>(x, m1, w1, b1, x00);

  // Layer 2 (WMMA f32): x00 -> x01
  size_t lds2 = (size_t)(32 * 18 * 33 + 2 * 32 * 33) * sizeof(float);  // 84,480 B
  inrf2_kernel<<<512, 256, lds2, stream>>>(x00, m2, w2, b2, x01);

  // maxpool2: x01 -> x1
  int mp_total = 256 * 32 * 14 * 14;
  maxpool2_kernel<<<(mp_total + 255) / 256, 256, 0, stream>>>(x01, x1,
                                                              256 * 32, 28, 28);

  // Layer 3 (WMMA f32): x1 -> x02
  size_t lds3 = (size_t)(32 * 18 * 19 + 2 * 128 * 33) * sizeof(float); // 77,568 B
  inrf3_kernel<<<256, 256, lds3, stream>>>(x1, m3, w3, b3, x02);

  // maxpool2 + avgpool7 + FC: x02 -> logits
  tail_kernel<<<256, 128, 0, stream>>>(x02, fcw, fcb, logits);
}